// RecommendationModel_57801669869916
// MI455X (gfx1250) — compile-verified
//
#include <hip/hip_runtime.h>

typedef __attribute__((ext_vector_type(2))) float v2f;
typedef __attribute__((ext_vector_type(8))) float v8f;

#define ND 64
#define L2EPS 1e-12f

// ---------------------------------------------------------------------------
// Row L2 normalization: one wave (32 lanes) per 64-float row, float2 per lane.
// ---------------------------------------------------------------------------
__global__ __launch_bounds__(256) void k_l2norm_rows(const float* __restrict__ in,
                                                     float* __restrict__ out, int rows) {
    int gt   = blockIdx.x * blockDim.x + threadIdx.x;
    int r    = gt >> 5;
    int lane = gt & 31;
    if (r >= rows) return;
    float2 v = ((const float2*)(in + (size_t)r * ND))[lane];
    float p  = v.x * v.x + v.y * v.y;
    p += __shfl_xor(p, 1, 32);
    p += __shfl_xor(p, 2, 32);
    p += __shfl_xor(p, 4, 32);
    p += __shfl_xor(p, 8, 32);
    p += __shfl_xor(p, 16, 32);
    float s = 1.0f / fmaxf(sqrtf(p), L2EPS);
    float2 o; o.x = v.x * s; o.y = v.y * s;
    ((float2*)(out + (size_t)r * ND))[lane] = o;
}

// ---------------------------------------------------------------------------
// Zero scratch (agg + cnt contiguous).
// ---------------------------------------------------------------------------
__global__ __launch_bounds__(256) void k_zero(float* __restrict__ p, int n) {
    int i = blockIdx.x * blockDim.x + threadIdx.x;
    if (i < n) p[i] = 0.0f;
}

// ---------------------------------------------------------------------------
// Edge scatter: one wave per edge; lane carries 2 floats of the 64-float
// message. Atomic f32 adds resolve in L2 (the 25.6 MB feature slab and the
// 25.6 MB accumulator are both L2-resident on a 192 MB L2).
// ---------------------------------------------------------------------------
__global__ __launch_bounds__(256) void k_scatter(const float* __restrict__ x,
                                                 const long long* __restrict__ eidx,
                                                 float* __restrict__ agg,
                                                 float* __restrict__ cnt, int nedge) {
    int gt   = blockIdx.x * blockDim.x + threadIdx.x;
    int e    = gt >> 5;
    int lane = gt & 31;
    if (e >= nedge) return;
    int src = (int)eidx[e];
    int dst = (int)eidx[(size_t)nedge + e];
    float2 m = ((const float2*)(x + (size_t)src * ND))[lane];
    float* a = agg + (size_t)dst * ND + lane * 2;
    atomicAdd(a, m.x);
    atomicAdd(a + 1, m.y);
    if (lane == 0) atomicAdd(cnt + dst, 1.0f);
}

// ---------------------------------------------------------------------------
// Fused SAGE layer: out = l2norm(relu(mean @ Wl + b + x @ Wr)), in place.
// 256 threads = 8 waves; each wave owns a 16-row x 64-col output tile and
// runs 128 v_wmma_f32_16x16x4_f32 (2 GEMMs x 16 K-steps x 4 N-tiles).
//
// Weights are staged in LDS PAIR-SWIZZLED: sW2[p*64 + c] (float2) holds
// {W[2p][c], W[2p+1][c]}, so each B fragment is one aligned ds_load_b64
// that feeds the WMMA even-pair operand directly (no repack movs).
// ---------------------------------------------------------------------------
__global__ __launch_bounds__(256) void k_sage_wmma(const float* __restrict__ agg,
                                                   const float* __restrict__ cnt,
                                                   const float* __restrict__ Wl,
                                                   const float* __restrict__ bl,
                                                   const float* __restrict__ Wr,
                                                   float* __restrict__ xio, int n) {
    __shared__ __align__(16) float2 sWl2[ND * ND / 2];  // [p*64 + c] = {Wl[2p][c], Wl[2p+1][c]}
    __shared__ __align__(16) float2 sWr2[ND * ND / 2];
    __shared__ float sB[ND];

    for (int i = threadIdx.x; i < ND * ND / 2; i += blockDim.x) {
        int p = i >> 6;   // K-pair index 0..31
        int c = i & 63;   // column
        float2 a; a.x = Wl[(2 * p) * ND + c]; a.y = Wl[(2 * p + 1) * ND + c];
        sWl2[i] = a;
        float2 b; b.x = Wr[(2 * p) * ND + c]; b.y = Wr[(2 * p + 1) * ND + c];
        sWr2[i] = b;
    }
    if (threadIdx.x < ND) sB[threadIdx.x] = bl[threadIdx.x];
    __syncthreads();

    int wave = threadIdx.x >> 5;
    int lane = threadIdx.x & 31;
    int hh   = lane >> 4;   // lane half: selects K sub-pair / output row offset
    int l16  = lane & 15;
    int row0 = (blockIdx.x * 8 + wave) * 16;
    if (row0 >= n) return;  // N % 16 == 0, so tiles are always full

    int arow   = row0 + l16;
    float invc = 1.0f / fmaxf(cnt[arow], 1.0f);
    const float* Am = agg + (size_t)arow * ND;
    const float* Ax = xio + (size_t)arow * ND;

    v8f acc[4] = {};    // 4 column tiles of the 16x64 output strip

#pragma unroll
    for (int kk = 0; kk < 16; ++kk) {
        int kb = kk * 4 + hh * 2;   // A layout: v0 = K[kb], v1 = K[kb+1]
        int p  = kb >> 1;           // = 2*kk + hh : K-pair index into swizzled LDS
        float2 am2 = *(const float2*)(Am + kb);
        float2 ax2 = *(const float2*)(Ax + kb);
        v2f am; am[0] = am2.x * invc; am[1] = am2.y * invc;
        v2f ax; ax[0] = ax2.x;        ax[1] = ax2.y;
#pragma unroll
        for (int nt = 0; nt < 4; ++nt) {
            int col = nt * 16 + l16;
            float2 bm2 = sWl2[p * ND + col];
            float2 bx2 = sWr2[p * ND + col];
            v2f bm; bm[0] = bm2.x; bm[1] = bm2.y;
            v2f bx; bx[0] = bx2.x; bx[1] = bx2.y;
            acc[nt] = __builtin_amdgcn_wmma_f32_16x16x4_f32(
                false, am, false, bm, (short)0, acc[nt], false, false);
            acc[nt] = __builtin_amdgcn_wmma_f32_16x16x4_f32(
                false, ax, false, bx, (short)0, acc[nt], false, false);
        }
    }

    // bias + relu
#pragma unroll
    for (int nt = 0; nt < 4; ++nt) {
        float b = sB[nt * 16 + l16];
#pragma unroll
        for (int r = 0; r < 8; ++r)
            acc[nt][r] = fmaxf(acc[nt][r] + b, 0.0f);
    }

    // per-row L2 norm (row r+8*hh lives in lanes hh*16..hh*16+15) + store
#pragma unroll
    for (int r = 0; r < 8; ++r) {
        float p = 0.0f;
#pragma unroll
        for (int nt = 0; nt < 4; ++nt) p += acc[nt][r] * acc[nt][r];
        p += __shfl_xor(p, 1, 32);
        p += __shfl_xor(p, 2, 32);
        p += __shfl_xor(p, 4, 32);
        p += __shfl_xor(p, 8, 32);   // bit4 untouched: halves reduce separately
        float s   = 1.0f / fmaxf(sqrtf(p), L2EPS);
        int   row = row0 + r + hh * 8;
        float* o  = xio + (size_t)row * ND;
#pragma unroll
        for (int nt = 0; nt < 4; ++nt) o[nt * 16 + l16] = acc[nt][r] * s;
    }
}

// ---------------------------------------------------------------------------
// Driver. Inputs (dict order): x0,e0,x1,e1,x2,e2,Wl,bl,Wr.
// d_out = [x0_out | x1_out | x2_out] and doubles as the live feature state.
// d_ws  = per-type [agg(N*64) | cnt(N)]  (~78 MB).
// ---------------------------------------------------------------------------
extern "C" void kernel_launch(void* const* d_in, const int* in_sizes, int n_in,
                              void* d_out, int out_size, void* d_ws, size_t ws_size,
                              hipStream_t stream) {
    (void)in_sizes; (void)n_in; (void)out_size; (void)ws_size;
    const int N = 100000, NE = 800000;

    const float*     x[3] = {(const float*)d_in[0], (const float*)d_in[2], (const float*)d_in[4]};
    const long long* e[3] = {(const long long*)d_in[1], (const long long*)d_in[3], (const long long*)d_in[5]};
    const float* Wl = (const float*)d_in[6];
    const float* bl = (const float*)d_in[7];
    const float* Wr = (const float*)d_in[8];

    float* ws = (float*)d_ws;
    const size_t per = (size_t)N * ND + N;

    for (int k = 0; k < 3; ++k) {
        float* cur = (float*)d_out + (size_t)k * N * ND;
        float* agg = ws + (size_t)k * per;
        float* cnt = agg + (size_t)N * ND;

        k_l2norm_rows<<<(N + 7) / 8, 256, 0, stream>>>(x[k], cur, N);

        for (int l = 0; l < 3; ++l) {  // weights are per-type (Wl[k]), reused each layer
            int nz = N * ND + N;
            k_zero<<<(nz + 255) / 256, 256, 0, stream>>>(agg, nz);
            k_scatter<<<(NE * 32 + 255) / 256, 256, 0, stream>>>(cur, e[k], agg, cnt, NE);
            k_sage_wmma<<<(N / 16 + 7) / 8, 256, 0, stream>>>(
                agg, cnt, Wl + (size_t)k * ND * ND, bl + (size_t)k * ND,
                Wr + (size_t)k * ND * ND, cur, N);
        }
    }
}